// ForwardCASSIReal_38439957299668
// MI455X (gfx1250) — compile-verified
//
#include <hip/hip_runtime.h>
#include <cstdint>

// ---------------- problem constants (from reference) ----------------
constexpr int kB = 8;
constexpr int kM = 512;                       // height
constexpr int kW = 512;                       // width
constexpr int kL = 31;                        // spectral bands
constexpr int kWOUT = kW + kL - 1;            // 542
constexpr long long kNY = (long long)kB * kM * kWOUT;   // 2,220,032 floats (Y)
constexpr long long kNH = (long long)kM * kW * kL;      // 8,126,464 floats (H5 = H reshaped)

// ---------------- tiling ----------------
constexpr int kTW      = 184;                 // outputs per chunk (3*184 = 552 >= 542)
constexpr int kNCHUNK  = 3;
constexpr int kMAXCOLS = 216;                 // worst-case staged input columns (4-aligned start)
constexpr int kNDW     = kMAXCOLS * kL;       // 6696 dwords per staged array (mult of 4)
constexpr int kTHREADS = 192;                 // 6 waves (wave32)

// -------- CDNA5 async global->LDS data mover (ASYNCcnt-tracked) --------
__device__ __forceinline__ void async_load_b128_to_lds(const void* gptr, const void* lds_ptr) {
  // Generic pointer into __shared__ -> 32-bit LDS byte offset. The gfx1250
  // backend lowers flat->local addrspacecast as a truncation of the flat
  // address (LDS aperture keeps the offset in the low 32 bits), so this is
  // exactly equivalent to the compiler's own cast.
  unsigned lds_off = (unsigned)(uintptr_t)lds_ptr;
  unsigned long long ga = (unsigned long long)(uintptr_t)gptr;
  asm volatile("global_load_async_to_lds_b128 %0, %1, off"
               :: "v"(lds_off), "v"(ga)
               : "memory");
}

__device__ __forceinline__ void wait_async_zero() {
  asm volatile("s_wait_asynccnt 0x0" ::: "memory");
}

// ---------------- kernel 1: shift-and-sum + per-block max ----------------
__global__ __launch_bounds__(kTHREADS) void cassi_fwd_kernel(
    const float* __restrict__ X, const float* __restrict__ H,
    float* __restrict__ Yraw, float* __restrict__ blkmax) {
  __shared__ __align__(16) float lds[2 * kNDW];      // [X chunk | H chunk]  ~52.3 KB
  __shared__ float wmax[kTHREADS / 32];

  const int blk   = blockIdx.x;
  const int chunk = blk % kNCHUNK;
  const int row   = blk / kNCHUNK;                   // b*512 + m
  const int m     = row % kM;
  const int b     = row / kM;

  const int w0 = chunk * kTW;
  int c0 = w0 - (kL - 1); if (c0 < 0) c0 = 0;
  c0 &= ~3;                                          // 16B-aligned global start
  int c1 = w0 + kTW; if (c1 > kW) c1 = kW;           // both mult of 4 -> ndw mult of 4
  const int ndw  = (c1 - c0) * kL;
  const int nvec = ndw >> 2;

  const float4* gx = (const float4*)(X + ((size_t)((size_t)b * kM + m) * kW + c0) * kL);
  const float4* gh = (const float4*)(H + ((size_t)m * kW + c0) * kL);
  float4* lx = (float4*)&lds[0];
  float4* lh = (float4*)&lds[kNDW];

  // Perfectly coalesced async DMA into LDS; <= 18 outstanding per wave (cnt is 6b).
  for (int i = threadIdx.x; i < nvec; i += kTHREADS) {
    async_load_b128_to_lds(gx + i, lx + i);
    async_load_b128_to_lds(gh + i, lh + i);
  }
  wait_async_zero();
  __syncthreads();

  float acc = 0.0f;
  const int t = threadIdx.x;
  const int wout = w0 + t;
  if (t < kTW && wout < kWOUT) {
    int ilo = wout - (kW - 1); if (ilo < 0) ilo = 0;
    int ihi = wout;            if (ihi > kL - 1) ihi = kL - 1;
    const int base = (wout - c0) * kL;               // idx(i) = base - 30*i
    const float* sx = &lds[0];
    const float* sh = &lds[kNDW];
    for (int i = ilo; i <= ihi; ++i) {
      const int idx = base - (kL - 1) * i;           // lane stride 31 dwords: bank-conflict-free
      acc = fmaf(sx[idx], sh[idx], acc);
    }
    Yraw[(size_t)((size_t)b * kM + m) * kWOUT + wout] = acc;
  }

  // Block max (all values >= 0); one global write per block, no atomics.
  float v = acc;
  for (int o = 16; o > 0; o >>= 1) v = fmaxf(v, __shfl_xor(v, o, 32));
  if ((t & 31) == 0) wmax[t >> 5] = v;
  __syncthreads();
  if (t == 0) {
    float bm = wmax[0];
    for (int w = 1; w < kTHREADS / 32; ++w) bm = fmaxf(bm, wmax[w]);
    blkmax[blk] = bm;
  }
}

// ---------------- kernel 2: reduce block maxima -> global max ----------------
__global__ __launch_bounds__(1024) void reduce_max_kernel(
    const float* __restrict__ blkmax, int n, float* __restrict__ out_max) {
  __shared__ float s[32];
  float v = 0.0f;
  for (int i = threadIdx.x; i < n; i += 1024) v = fmaxf(v, blkmax[i]);
  for (int o = 16; o > 0; o >>= 1) v = fmaxf(v, __shfl_xor(v, o, 32));
  if ((threadIdx.x & 31) == 0) s[threadIdx.x >> 5] = v;
  __syncthreads();
  if (threadIdx.x < 32) {
    float w = s[threadIdx.x];
    for (int o = 16; o > 0; o >>= 1) w = fmaxf(w, __shfl_xor(w, o, 32));
    if (threadIdx.x == 0) out_max[0] = w;
  }
}

// ---------------- kernel 3: in-place normalize Y ----------------
__global__ __launch_bounds__(256) void scale_kernel(
    float* __restrict__ y, const float* __restrict__ gmax, int n4) {
  const float mx = gmax[0];
  const int i = blockIdx.x * 256 + threadIdx.x;
  if (i < n4) {
    float4 v = ((float4*)y)[i];
    v.x /= mx; v.y /= mx; v.z /= mx; v.w /= mx;
    ((float4*)y)[i] = v;
  }
}

// ---------------- host entry ----------------
extern "C" void kernel_launch(void* const* d_in, const int* in_sizes, int n_in,
                              void* d_out, int out_size, void* d_ws, size_t ws_size,
                              hipStream_t stream) {
  (void)in_sizes; (void)n_in; (void)out_size; (void)ws_size;
  const float* X = (const float*)d_in[0];
  const float* H = (const float*)d_in[1];
  float* Yout  = (float*)d_out;                 // first kNY floats: Y
  float* H5out = (float*)d_out + kNY;           // next kNH floats: H5 (flat copy of H)

  float* wsf     = (float*)d_ws;
  float* maxslot = wsf;                         // 1 float
  float* blkmax  = wsf + 16;                    // kB*kM*kNCHUNK floats (~48 KB of ws)

  const int nblk = kB * kM * kNCHUNK;           // 12288 blocks
  cassi_fwd_kernel<<<nblk, kTHREADS, 0, stream>>>(X, H, Yout, blkmax);
  reduce_max_kernel<<<1, 1024, 0, stream>>>(blkmax, nblk, maxslot);

  const int n4 = (int)(kNY / 4);                // kNY divisible by 4
  scale_kernel<<<(n4 + 255) / 256, 256, 0, stream>>>(Yout, maxslot, n4);

  hipMemcpyAsync(H5out, H, (size_t)kNH * sizeof(float),
                 hipMemcpyDeviceToDevice, stream);
}